// VariationalQuantumCircuit_46806553591873
// MI455X (gfx1250) — compile-verified
//
#include <hip/hip_runtime.h>
#include <hip/hip_bf16.h>

typedef float v2f __attribute__((ext_vector_type(2)));
typedef float v8f __attribute__((ext_vector_type(8)));

#define N_QUBITS 4
#define N_LAYERS 6
#define DIM 16   // 2^N_QUBITS

// ---------------------------------------------------------------------------
// Kernel 1: build the fixed 16x16 circuit unitary U from the shared rotation
// params. Thread c evolves basis column |c>. Writes U^T (row-major, [j][i])
// real part to ws[0..255], imag part to ws[256..511].
// ---------------------------------------------------------------------------
__global__ void vqc_build_unitary(const float* __restrict__ rot,
                                  float* __restrict__ ws) {
    __shared__ float sr[DIM][DIM];
    __shared__ float si[DIM][DIM];
    const int c = threadIdx.x;
    if (c >= DIM) return;

    for (int i = 0; i < DIM; ++i) { sr[c][i] = (i == c) ? 1.0f : 0.0f; si[c][i] = 0.0f; }

    for (int l = 0; l < N_LAYERS; ++l) {
        // Rot(phi, theta, omega) = RZ(omega) RY(theta) RZ(phi) on qubit q
        for (int q = 0; q < N_QUBITS; ++q) {
            const int base = (l * N_QUBITS + q) * 3;
            const float phi = rot[base + 0];
            const float th  = rot[base + 1];
            const float om  = rot[base + 2];
            const float ct = __cosf(0.5f * th), st = __sinf(0.5f * th);
            const float a  = 0.5f * (phi + om);
            const float bb = 0.5f * (phi - om);
            const float ca = __cosf(a),  sa = __sinf(a);
            const float cb = __cosf(bb), sb = __sinf(bb);
            // U00 = e^{-ia} ct ; U01 = -e^{ib} st ; U10 = e^{-ib} st ; U11 = e^{ia} ct
            const float u00r =  ca * ct, u00i = -sa * ct;
            const float u01r = -cb * st, u01i = -sb * st;
            const float u10r =  cb * st, u10i = -sb * st;
            const float u11r =  ca * ct, u11i =  sa * ct;
            const int mask = 1 << (3 - q);
            for (int i = 0; i < DIM; ++i) {
                if (i & mask) continue;
                const int i1 = i | mask;
                const float v0r = sr[c][i],  v0i = si[c][i];
                const float v1r = sr[c][i1], v1i = si[c][i1];
                sr[c][i]  = u00r * v0r - u00i * v0i + u01r * v1r - u01i * v1i;
                si[c][i]  = u00r * v0i + u00i * v0r + u01r * v1i + u01i * v1r;
                sr[c][i1] = u10r * v0r - u10i * v0i + u11r * v1r - u11i * v1i;
                si[c][i1] = u10r * v0i + u10i * v0r + u11r * v1i + u11i * v1r;
            }
        }
        // Ring of CNOTs, range r = l % 3 + 1, applied sequentially q = 0..3
        const int rr = l % (N_QUBITS - 1) + 1;
        for (int cq = 0; cq < N_QUBITS; ++cq) {
            const int tq = (cq + rr) & 3;
            const int cmask = 1 << (3 - cq);
            const int tmask = 1 << (3 - tq);
            for (int i = 0; i < DIM; ++i) {
                if ((i & cmask) && !(i & tmask)) {
                    const int i1 = i | tmask;
                    float t;
                    t = sr[c][i]; sr[c][i] = sr[c][i1]; sr[c][i1] = t;
                    t = si[c][i]; si[c][i] = si[c][i1]; si[c][i1] = t;
                }
            }
        }
    }

    // Thread c holds column c of U: U[i][c].  U^T[j][i] = U[i][j] -> row j=c.
    for (int i = 0; i < DIM; ++i) {
        ws[c * DIM + i]       = sr[c][i];
        ws[256 + c * DIM + i] = si[c][i];
    }
}

// ---------------------------------------------------------------------------
// Kernel 2: per wave, a tile of 16 samples.
//   Psi[m, j]   built from RX embedding (complex, tensor-product closed form)
//   Phi = Psi * U^T  via v_wmma_f32_16x16x4_f32 (complex = 4 real matmuls,
//                     K=16 accumulated in 4 chunks of 4)
//   <Z_q> via a 4-stage Walsh-Hadamard butterfly over the 16-lane half:
//   after the WHT, lane n holds Sum_m p_m (-1)^{popc(n&m)}; the outputs are
//   the coefficients at n = 8,4,2,1 (q = 0..3).
// ---------------------------------------------------------------------------
__global__ __launch_bounds__(256) void vqc_forward(const float* __restrict__ x,
                                                   const float* __restrict__ ws,
                                                   float* __restrict__ out,
                                                   int Btot) {
    // Stage U^T (512 floats) into LDS once per block.
    __shared__ float lUr[256];
    __shared__ float lUi[256];
    {
        const int t = threadIdx.x; // 256 threads
        lUr[t] = ws[t];
        lUi[t] = ws[256 + t];
    }
    __syncthreads();

    const int lane = threadIdx.x & 31;
    const int half = lane >> 4;      // 0: K offsets {0,1}, 1: K offsets {2,3}
    const int m    = lane & 15;      // sample row within tile / N column
    const int wave = threadIdx.x >> 5;
    const int tile = (blockIdx.x * (blockDim.x >> 5) + wave) * DIM;

    int b = tile + m;
    if (b >= Btot) b = Btot - 1;     // clamp loads; stores are guarded below

    // Embedding amplitudes: psi_j = (-i)^popc(j) * prod_q (bit ? sin : cos)(x_q/2)
    float cq[4], sq[4];
    #pragma unroll
    for (int q = 0; q < 4; ++q) {
        __sincosf(0.5f * x[b * 4 + q], &sq[q], &cq[q]);
    }

    // A-operand values for the 8 K-slots this lane owns (ISA 16x4 f32 A layout:
    // chunk t, vgpr u -> K = 4t + 2*half + u), plus the B-operand (U^T) values.
    float Are[8], Aim[8], AimN[8], Brv[8], Biv[8];
    #pragma unroll
    for (int t = 0; t < 4; ++t) {
        #pragma unroll
        for (int u = 0; u < 2; ++u) {
            const int j = 4 * t + 2 * half + u;   // K index (input basis state)
            const float f0 = ((j >> 3) & 1) ? sq[0] : cq[0];
            const float f1 = ((j >> 2) & 1) ? sq[1] : cq[1];
            const float f2 = ((j >> 1) & 1) ? sq[2] : cq[2];
            const float f3 = ((j >> 0) & 1) ? sq[3] : cq[3];
            const float amp = f0 * f1 * f2 * f3;
            const int p = __popc(j) & 3;          // (-i)^p phase
            const float re = (p == 0) ? amp : ((p == 2) ? -amp : 0.0f);
            const float im = (p == 1) ? -amp : ((p == 3) ? amp : 0.0f);
            const int s = 2 * t + u;
            Are[s] = re; Aim[s] = im; AimN[s] = -im;
            Brv[s] = lUr[j * DIM + m];            // B[k][n] = U^T[k][n]
            Biv[s] = lUi[j * DIM + m];
        }
    }

    // Complex GEMM: Phi_r = Psi_r*Br - Psi_i*Bi ; Phi_i = Psi_r*Bi + Psi_i*Br
    v8f dr = {};
    v8f di = {};
    #pragma unroll
    for (int t = 0; t < 4; ++t) {
        v2f ar  = { Are[2 * t],  Are[2 * t + 1]  };
        v2f ai  = { Aim[2 * t],  Aim[2 * t + 1]  };
        v2f ain = { AimN[2 * t], AimN[2 * t + 1] };
        v2f br  = { Brv[2 * t],  Brv[2 * t + 1]  };
        v2f bi  = { Biv[2 * t],  Biv[2 * t + 1]  };
        dr = __builtin_amdgcn_wmma_f32_16x16x4_f32(false, ar,  false, br, (short)0, dr, false, false);
        dr = __builtin_amdgcn_wmma_f32_16x16x4_f32(false, ain, false, bi, (short)0, dr, false, false);
        di = __builtin_amdgcn_wmma_f32_16x16x4_f32(false, ar,  false, bi, (short)0, di, false, false);
        di = __builtin_amdgcn_wmma_f32_16x16x4_f32(false, ai,  false, br, (short)0, di, false, false);
    }

    // Which output column this lane would own after the WHT: lanes with a
    // single set bit n hold <Z_q> with q = 4 - ffs(n) (n=8->q0 ... n=1->q3).
    const bool isPow2 = (m != 0) && ((m & (m - 1)) == 0);
    const int  qOut   = 4 - __ffs(m);              // valid only when isPow2

    // Epilogue: probs -> Walsh-Hadamard -> <Z_q>.
    // C/D layout: vgpr r holds (M = r + 8*half, N = lane%16).
    #pragma unroll
    for (int r = 0; r < 8; ++r) {
        const float pr = dr[r], pi = di[r];
        float v = pr * pr + pi * pi;               // prob of state n=m, sample M
        #pragma unroll
        for (int mask = 1; mask <= 8; mask <<= 1) {
            const float w = __shfl_xor(v, mask, 32);
            v = (m & mask) ? (w - v) : (v + w);    // signed butterfly stage
        }
        const int bo = tile + r + 8 * half;        // sample index for this row
        if (isPow2 && bo < Btot) out[bo * 4 + qOut] = v;
    }
}

// ---------------------------------------------------------------------------
extern "C" void kernel_launch(void* const* d_in, const int* in_sizes, int n_in,
                              void* d_out, int out_size, void* d_ws, size_t ws_size,
                              hipStream_t stream) {
    const float* inputs    = (const float*)d_in[0];  // [B, 4]
    const float* rotations = (const float*)d_in[1];  // [6, 4, 3]
    float* out = (float*)d_out;                      // [B, 4]
    float* ws  = (float*)d_ws;                       // 512 floats used

    const int Btot = in_sizes[0] / N_QUBITS;

    vqc_build_unitary<<<1, 16, 0, stream>>>(rotations, ws);

    const int samplesPerBlock = 128;                 // 8 waves * 16 samples
    const int grid = (Btot + samplesPerBlock - 1) / samplesPerBlock;
    vqc_forward<<<grid, 256, 0, stream>>>(inputs, ws, out, Btot);
}